// TripleLayerBioPINN_72782515798419
// MI455X (gfx1250) — compile-verified
//
#include <hip/hip_runtime.h>
#include <hip/hip_bf16.h>

// CDNA5 (gfx1250, wave32) fused kernel:
//  phase 1: per-thread register-resident ODE simulation (one cell per lane)
//  phase 2: block-level 96->256 linear projection via V_WMMA_F32_16X16X4_F32
//           (A from LDS-staged phys vectors, B from lin_w, C seeded with bias)

typedef __attribute__((ext_vector_type(2))) float v2f;
typedef __attribute__((ext_vector_type(8))) float v8f;

#define BLK 8
#define D_MODEL 256
#define FLAT_DIM 96            // 8*4 (h) + 8*8 (W)
#define CELLS_PER_WG 256       // 8 wave32s
#define LDS_STRIDE 98          // even (float2-aligned), spreads LDS banks

__device__ __forceinline__ float clamp01(float x) {
    return fminf(fmaxf(x, 0.0f), 1.0f);
}

__global__ __launch_bounds__(CELLS_PER_WG)
void biopinn_fused_kernel(const int*   __restrict__ bits,    // [N,8,8] int32
                          const float* __restrict__ stim_in, // [N]
                          const float* __restrict__ Winit,   // [N,8,8]
                          const float* __restrict__ lin_w,   // [256,96]
                          const float* __restrict__ lin_b,   // [256]
                          const int*   __restrict__ steps_p, // [1]
                          float* __restrict__ out_emb,       // [N,256]
                          float* __restrict__ out_h,         // [N,32]
                          float* __restrict__ out_W)         // [N,64]
{
    __shared__ float lds_phys[CELLS_PER_WG * LDS_STRIDE];

    const int tid  = threadIdx.x;
    const size_t cell = (size_t)blockIdx.x * CELLS_PER_WG + tid;
    const int steps = steps_p[0];

    float h[BLK][4];
    float W[BLK][BLK];

    // ---- decode bits -> EPGL in {0, 1/3, 2/3, 1} ----
    {
        const int* bp = bits + cell * 64;
        #pragma unroll
        for (int i = 0; i < BLK; ++i) {
            int4 lo = *(const int4*)(bp + i * 8);
            int4 hi = *(const int4*)(bp + i * 8 + 4);
            h[i][0] = (float)(lo.x * 2 + lo.y) * (1.0f / 3.0f);
            h[i][1] = (float)(lo.z * 2 + lo.w) * (1.0f / 3.0f);
            h[i][2] = (float)(hi.x * 2 + hi.y) * (1.0f / 3.0f);
            h[i][3] = (float)(hi.z * 2 + hi.w) * (1.0f / 3.0f);
        }
    }
    const float stim = stim_in[cell];
    {
        const float* wp = Winit + cell * 64;
        #pragma unroll
        for (int i = 0; i < BLK; ++i) {
            float4 a = *(const float4*)(wp + i * 8);
            float4 b = *(const float4*)(wp + i * 8 + 4);
            W[i][0] = a.x; W[i][1] = a.y; W[i][2] = a.z; W[i][3] = a.w;
            W[i][4] = b.x; W[i][5] = b.y; W[i][6] = b.z; W[i][7] = b.w;
            W[i][i] = 0.0f;   // * (1 - eye)
        }
    }

    // ---- internal dynamics ----
    for (int s = 0; s < steps; ++s) {
        float neigh[BLK][4];
        #pragma unroll
        for (int i = 0; i < BLK; ++i) {
            float tw = 1e-8f;
            float a0 = 0.f, a1 = 0.f, a2 = 0.f, a3 = 0.f;
            #pragma unroll
            for (int k = 0; k < BLK; ++k) {
                float w = W[i][k];
                tw += w;
                a0 += w * h[k][0];
                a1 += w * h[k][1];
                a2 += w * h[k][2];
                a3 += w * h[k][3];
            }
            neigh[i][0] = a0 / tw; neigh[i][1] = a1 / tw;
            neigh[i][2] = a2 / tw; neigh[i][3] = a3 / tw;
        }
        #pragma unroll
        for (int i = 0; i < BLK; ++i) {
            float E = h[i][0], P = h[i][1], G = h[i][2], L = h[i][3];
            float En = neigh[i][0], Pn = neigh[i][1], Gn = neigh[i][2], Ln = neigh[i][3];
            float E_new = clamp01(E + 0.3f * stim - 0.4f * P - 0.2f * G);
            float P_new = clamp01(P + 0.5f * stim + 0.3f * (Pn - P) - 0.2f * E);
            float G_new = clamp01(G + 0.4f * E * (1.0f - P) + 0.2f * (Gn - G) - 0.3f * P);
            float good  = 0.5f * En + 0.5f * Gn;
            float L_new = clamp01(L + 0.4f * good + 0.3f * (Ln - L) - 0.3f * P);
            h[i][0] = E_new; h[i][1] = P_new; h[i][2] = G_new; h[i][3] = L_new;
        }
        #pragma unroll
        for (int i = 0; i < BLK; ++i) {
            #pragma unroll
            for (int k = 0; k < BLK; ++k) {
                float d0 = h[i][0] - h[k][0];
                float d1 = h[i][1] - h[k][1];
                float d2 = h[i][2] - h[k][2];
                float d3 = h[i][3] - h[k][3];
                float dist = sqrtf(d0 * d0 + d1 * d1 + d2 * d2 + d3 * d3); // sqrt(0)=0: safe pdist
                float mutual = 0.5f * (h[i][3] + h[k][3]);
                float w = W[i][k];
                float wn = clamp01(w + 0.1f * mutual * dist - 0.05f * w);
                W[i][k] = (i == k) ? 0.0f : wn;                            // * (1 - eye)
            }
        }
    }

    // ---- write h, W outputs ----
    {
        float* hp = out_h + cell * 32;
        #pragma unroll
        for (int i = 0; i < BLK; ++i) {
            float4 v = {h[i][0], h[i][1], h[i][2], h[i][3]};
            *(float4*)(hp + i * 4) = v;
        }
        float* wp = out_W + cell * 64;
        #pragma unroll
        for (int i = 0; i < BLK; ++i) {
            float4 a = {W[i][0], W[i][1], W[i][2], W[i][3]};
            float4 b = {W[i][4], W[i][5], W[i][6], W[i][7]};
            *(float4*)(wp + i * 8)     = a;
            *(float4*)(wp + i * 8 + 4) = b;
        }
    }

    // ---- stage phys = [h(32) | W(64)] into LDS ----
    {
        float* p = lds_phys + tid * LDS_STRIDE;
        #pragma unroll
        for (int i = 0; i < BLK; ++i)
            #pragma unroll
            for (int j = 0; j < 4; ++j)
                p[i * 4 + j] = h[i][j];
        #pragma unroll
        for (int i = 0; i < BLK; ++i)
            #pragma unroll
            for (int k = 0; k < BLK; ++k)
                p[32 + i * 8 + k] = W[i][k];
    }
    __syncthreads();

    // ---- 96 -> 256 projection with V_WMMA_F32_16X16X4_F32 ----
    // out[m, n] = sum_f phys[m, f] * lin_w[n, f] + lin_b[n]
    const int lane  = tid & 31;
    const int wave  = tid >> 5;
    const int nrow  = lane & 15;   // M for A-frag, N for B/C-frag
    const int khalf = lane >> 4;   // K half: lanes 0-15 -> K{0,1}, 16-31 -> K{2,3}

    for (int mt = wave * 2; mt < wave * 2 + 2; ++mt) {           // 16 M-tiles / 8 waves
        const size_t cell0 = (size_t)blockIdx.x * CELLS_PER_WG + mt * 16;
        for (int nt = 0; nt < 16; ++nt) {                        // 256 / 16 N-tiles
            const int n0 = nt * 16;
            const float bias = lin_b[n0 + nrow];
            v8f acc = {bias, bias, bias, bias, bias, bias, bias, bias};
            #pragma unroll
            for (int kc = 0; kc < FLAT_DIM / 4; ++kc) {          // 24 K-chunks of 4
                const int kbase = kc * 4 + khalf * 2;            // even -> 8B aligned
                v2f a = *(const v2f*)(lds_phys + (mt * 16 + nrow) * LDS_STRIDE + kbase);
                v2f b = *(const v2f*)(lin_w + (size_t)(n0 + nrow) * FLAT_DIM + kbase);
                acc = __builtin_amdgcn_wmma_f32_16x16x4_f32(
                          /*neg_a=*/false, a, /*neg_b=*/false, b,
                          /*c_mod=*/(short)0, acc,
                          /*reuse_a=*/false, /*reuse_b=*/false);
            }
            // C/D layout: VGPR r, lane l -> M = r + (l>=16)*8, N = l&15
            float* op = out_emb + cell0 * D_MODEL + n0;
            #pragma unroll
            for (int r = 0; r < 8; ++r) {
                const int M = r + khalf * 8;
                op[(size_t)M * D_MODEL + nrow] = acc[r];
            }
        }
    }
}

extern "C" void kernel_launch(void* const* d_in, const int* in_sizes, int n_in,
                              void* d_out, int out_size, void* d_ws, size_t ws_size,
                              hipStream_t stream) {
    const int*   bits  = (const int*)d_in[0];   // bit_tensor  [B,S,8,8] int32
    const float* stim  = (const float*)d_in[1]; // macro_stimulus [B,S]
    const float* Winit = (const float*)d_in[2]; // W_init [B,S,8,8]
    const float* lin_w = (const float*)d_in[3]; // [256,96]
    const float* lin_b = (const float*)d_in[4]; // [256]
    const int*   steps = (const int*)d_in[5];   // internal_steps (scalar on device)

    const int ncells = in_sizes[0] / 64;        // B*S = 131072

    float* out_emb = (float*)d_out;                         // [N,256]
    float* out_h   = out_emb + (size_t)ncells * D_MODEL;    // [N,32]
    float* out_W   = out_h   + (size_t)ncells * 32;         // [N,64]

    const int grid = ncells / CELLS_PER_WG;     // 512 (exact for fixed shapes)
    hipLaunchKernelGGL(biopinn_fused_kernel, dim3(grid), dim3(CELLS_PER_WG), 0, stream,
                       bits, stim, Winit, lin_w, lin_b, steps,
                       out_emb, out_h, out_W);
}